// GlobalAverageBlock_49555332661495
// MI455X (gfx1250) — compile-verified
//
#include <hip/hip_runtime.h>

// Segmented mean: out[s,d] = (1/len[s]) * sum_{n: seg[n]==s} x[n,d]
// N=1048576 rows, B=64 segments (contiguous, sorted, each >=8192 rows), D=64.
//
// Memory-bound: 260MB @ 23.3TB/s ~= 11us floor. Accumulation runs on the
// matrix pipe via V_WMMA_F32_16X16X4_F32 with an exact f32 one-hot selector,
// dual accumulators for back-to-back XDL issue, and unguarded speculative
// global_prefetch_b8.

#define NROWS   1048576
#define BSEG    64
#define DFEAT   64
#define CHUNK   2048     // rows per workgroup; a chunk spans <= 2 segments
#define THREADS 256      // 8 wave32s: 4 feature tiles x 2 row halves
#define PF_ROWS 64       // prefetch distance in rows

typedef float v2f __attribute__((ext_vector_type(2)));
typedef float v8f __attribute__((ext_vector_type(8)));

__global__ void gab_zero_ws(float* __restrict__ ws) {
    int i = blockIdx.x * blockDim.x + threadIdx.x;
    if (i < BSEG * DFEAT) ws[i] = 0.0f;
}

__global__ __launch_bounds__(THREADS)
void gab_seg_sum_wmma(const float* __restrict__ x,
                      const int*   __restrict__ seg,
                      float*       __restrict__ ws) {
    const int tid  = threadIdx.x;
    const int lane = tid & 31;
    const int wave = tid >> 5;
    const int ft   = wave & 3;    // feature tile: columns ft*16 .. ft*16+15
    const int rh   = wave >> 2;   // row half of the chunk

    const int chunk_start = blockIdx.x * CHUNK;
    const int row_base    = chunk_start + rh * (CHUNK / 2);

    // Uniform per block; all rows of the chunk fall in [seg_base, seg_base+15]
    // because min segment length (8192) >= CHUNK => at most one boundary.
    const int seg_base = seg[chunk_start];

    const int m  = lane & 15;          // A-matrix M (segment within window)
    const int kh = (lane >> 4) << 1;   // K half: lanes 0-15 -> K={0,1}, 16-31 -> K={2,3}
    const int f  = ft * 16 + m;        // feature column owned by this lane (B/C N index)

    // Incrementally-advanced pointers: no per-iteration 64-bit address math.
    const float* px  = x   + (size_t)(row_base + kh) * DFEAT + f;
    const int*   ps  = seg + row_base + kh;
    const float* ppf = px  + (size_t)PF_ROWS * DFEAT;

    v8f c0 = {};  // 16 segs x 16 feats accumulator tiles (dual for XDL ILP)
    v8f c1 = {};

    // 8 rows (two 4-row K-groups -> two independent WMMAs) per iteration.
    for (int it = 0; it < (CHUNK / 2) / 8; ++it) {
        // Selector A[m][k] = 1.0f iff row belongs to segment seg_base+m.
        // ISA 32-bit A 16x4 layout: lanes 0-15 M=lane hold K=0 (v0), K=1 (v1);
        // lanes 16-31 M=lane-16 hold K=2 (v0), K=3 (v1).
        const int s0 = ps[0];
        const int s1 = ps[1];
        const int s2 = ps[4];
        const int s3 = ps[5];

        v2f a0, a1, b0, b1;
        a0.x = (s0 - seg_base == m) ? 1.0f : 0.0f;
        a0.y = (s1 - seg_base == m) ? 1.0f : 0.0f;
        a1.x = (s2 - seg_base == m) ? 1.0f : 0.0f;
        a1.y = (s3 - seg_base == m) ? 1.0f : 0.0f;

        // B 4x16 tile of x: row K, features ft*16+(lane&15), K split across
        // lane halves mirroring A's layout.
        b0.x = px[0];
        b0.y = px[DFEAT];
        b1.x = px[4 * DFEAT];
        b1.y = px[5 * DFEAT];

        // Speculative prefetch of rows PF_ROWS ahead (no guard needed).
        __builtin_prefetch(ppf, 0, 1);

        // D = A x B + C  (16x16x4 f32, exact accumulation), independent chains.
        c0 = __builtin_amdgcn_wmma_f32_16x16x4_f32(
                false, a0, false, b0, (short)0, c0, false, false);
        c1 = __builtin_amdgcn_wmma_f32_16x16x4_f32(
                false, a1, false, b1, (short)0, c1, false, false);

        px  += 8 * DFEAT;
        ps  += 8;
        ppf += 8 * DFEAT;
    }

    // C/D layout: VGPR r -> M = r (lanes 0-15) / r+8 (lanes 16-31), N = lane&15.
    const int mofs = (lane >> 4) * 8;
    #pragma unroll
    for (int r = 0; r < 8; ++r) {
        float v = c0[r] + c1[r];
        if (v != 0.0f) {                       // most of the window is empty
            int sgi = seg_base + r + mofs;
            if (sgi < BSEG)
                atomicAdd(&ws[sgi * DFEAT + f], v);   // global_atomic_add_f32
        }
    }
}

__global__ void gab_finalize(const float* __restrict__ ws,
                             const int*   __restrict__ lengths,
                             float*       __restrict__ out) {
    int i = blockIdx.x * blockDim.x + threadIdx.x;
    if (i < BSEG * DFEAT) {
        out[i] = ws[i] / (float)lengths[i / DFEAT];
    }
}

extern "C" void kernel_launch(void* const* d_in, const int* in_sizes, int n_in,
                              void* d_out, int out_size, void* d_ws, size_t ws_size,
                              hipStream_t stream) {
    const float* x       = (const float*)d_in[0];
    const int*   seg     = (const int*)  d_in[1];
    const int*   lengths = (const int*)  d_in[2];
    float*       out     = (float*)d_out;
    float*       ws      = (float*)d_ws;   // [BSEG*DFEAT] f32 accumulators

    gab_zero_ws<<<(BSEG * DFEAT + 255) / 256, 256, 0, stream>>>(ws);
    gab_seg_sum_wmma<<<NROWS / CHUNK, THREADS, 0, stream>>>(x, seg, ws);
    gab_finalize<<<(BSEG * DFEAT + 255) / 256, 256, 0, stream>>>(ws, lengths, out);
}